// CausalSelfAttention_29102698398197
// MI455X (gfx1250) — compile-verified
//
#include <hip/hip_runtime.h>
#include <hip/hip_bf16.h>

typedef __attribute__((ext_vector_type(2))) float v2f;
typedef __attribute__((ext_vector_type(8))) float v8f;

// ---------------------------------------------------------------------------
// D = A(16x4 f32) * B(4x16 f32) + C(16x16 f32), exact fp32 math on the
// CDNA5 matrix core.  A/B = 2 VGPRs/lane, C/D = 8 VGPRs/lane (wave32).
// ---------------------------------------------------------------------------
__device__ __forceinline__ v8f wmma_f32(v2f a, v2f b, v8f c) {
  return __builtin_amdgcn_wmma_f32_16x16x4_f32(
      /*neg_a=*/false, a, /*neg_b=*/false, b,
      /*c_mod=*/(short)0, c, /*reuse_a=*/false, /*reuse_b=*/false);
}

__device__ __forceinline__ v8f v8f_zero() {
  v8f z;
#pragma unroll
  for (int i = 0; i < 8; ++i) z[i] = 0.0f;
  return z;
}

// ---------------------------------------------------------------------------
// C[M,N] = A[M,K] @ B[K,N], row-major fp32.
// 256 threads = 8 waves; block tile 128x256; wave tile 64x64 (4x4 WMMA tiles).
// Requires M%128==0, N%256==0, K%4==0 (true for all calls below).
// ---------------------------------------------------------------------------
__global__ __launch_bounds__(256) void gemm_wmma_f32(
    const float* __restrict__ A, const float* __restrict__ B,
    float* __restrict__ C, int M, int N, int K) {
  const int lane = threadIdx.x & 31;
  const int wave = threadIdx.x >> 5;      // 0..7
  const int wm = wave >> 2;               // 0..1 (M)
  const int wn = wave & 3;                // 0..3 (N)
  const int m0 = blockIdx.y * 128 + wm * 64;
  const int n0 = blockIdx.x * 256 + wn * 64;

  const int half = lane >> 4;             // 0 or 1
  const int l16  = lane & 15;
  const int kb   = half * 2;              // k sub-offset {0,2}

  v8f acc[4][4];
#pragma unroll
  for (int i = 0; i < 4; ++i)
#pragma unroll
    for (int j = 0; j < 4; ++j) acc[i][j] = v8f_zero();

  for (int k = 0; k < K; k += 4) {
    v2f afrag[4];
#pragma unroll
    for (int mt = 0; mt < 4; ++mt) {
      const float* ap = A + (size_t)(m0 + mt * 16 + l16) * K + k + kb;
      afrag[mt] = *(const v2f*)ap;        // A[m][k+kb], A[m][k+kb+1]
    }
    v2f bfrag[4];
#pragma unroll
    for (int nt = 0; nt < 4; ++nt) {
      const float* bp = B + (size_t)(k + kb) * N + n0 + nt * 16 + l16;
      v2f bv;
      bv.x = bp[0];                       // B[k+kb  ][n]
      bv.y = bp[N];                       // B[k+kb+1][n]
      bfrag[nt] = bv;
    }
#pragma unroll
    for (int mt = 0; mt < 4; ++mt)
#pragma unroll
      for (int nt = 0; nt < 4; ++nt)
        acc[mt][nt] = wmma_f32(afrag[mt], bfrag[nt], acc[mt][nt]);
  }

#pragma unroll
  for (int mt = 0; mt < 4; ++mt)
#pragma unroll
    for (int nt = 0; nt < 4; ++nt)
#pragma unroll
      for (int r = 0; r < 8; ++r) {
        const int row = m0 + mt * 16 + r + half * 8;
        const int col = n0 + nt * 16 + l16;
        C[(size_t)row * N + col] = acc[mt][nt][r];
      }
}

// ---------------------------------------------------------------------------
// Flash attention, causal, with RoPE fused into the Q/K fragment loads.
// One wave handles one (b, h, 16-row query block); 8 waves per block.
// qkv layout: [B, S, 3E] with q at col h*D+d, k at E+h*D+d, v at 2E+h*D+d.
// freqs: [S, D/2, 2] (cos, sin).  out: [B, S, E] at col h*D+d.
// ---------------------------------------------------------------------------
#define AT_S 2048
#define AT_E 2048
#define AT_H 16
#define AT_D 128
#define AT_SCALE 0.08838834764831845f   // 1/sqrt(128)

__global__ __launch_bounds__(256) void attn_flash_wmma(
    const float* __restrict__ qkv, const float* __restrict__ freqs,
    float* __restrict__ out) {
  __shared__ float plds[8][16][17];      // P tile staging, padded rows

  const int lane = threadIdx.x & 31;
  const int wv   = threadIdx.x >> 5;
  const int wid  = blockIdx.x * 8 + wv;
  const int S16  = AT_S / 16;            // 128 query blocks per (b,h)
  const int qb   = wid % S16;
  const int h    = (wid / S16) % AT_H;
  const int b    = wid / (S16 * AT_H);
  const int q0   = qb * 16;

  const int half = lane >> 4;
  const int l16  = lane & 15;
  const int kb   = half * 2;

  const size_t rs = 3 * (size_t)AT_E;    // qkv row stride (floats)
  const float* qbase = qkv + (size_t)b * AT_S * rs + (size_t)h * AT_D;
  const float* kbase = qbase + AT_E;
  const float* vbase = qbase + 2 * AT_E;

  // ---- load Q tile (16x128) into 32 A-fragments, RoPE + scale fused ----
  v2f qfrag[32];
  {
    const int srow = q0 + l16;           // A-layout: m = lane&15
    const float* qrow = qbase + (size_t)srow * rs;
    const float* frow = freqs + (size_t)srow * AT_D;   // 64 pairs * 2
#pragma unroll
    for (int t = 0; t < 32; ++t) {
      const int d = t * 4 + kb;          // even -> rotary pair (d, d+1)
      v2f q = *(const v2f*)(qrow + d);
      const float c = frow[d];           // frow[2*(d/2)]
      const float s = frow[d + 1];
      v2f r;
      r.x = (q.x * c - q.y * s) * AT_SCALE;
      r.y = (q.y * c + q.x * s) * AT_SCALE;
      qfrag[t] = r;
    }
  }

  v8f acc[8];
#pragma unroll
  for (int nt = 0; nt < 8; ++nt) acc[nt] = v8f_zero();
  float rowmax[8], rowsum[8];
#pragma unroll
  for (int r = 0; r < 8; ++r) { rowmax[r] = -3.4e38f; rowsum[r] = 0.0f; }

  for (int key0 = 0; key0 <= q0; key0 += 16) {
    // ---- scores = Q @ K^T  (32 WMMAs over D=128), RoPE on K fused ----
    v8f sc = v8f_zero();
    const int srow = key0 + l16;         // B-layout: n = lane&15
    const float* krow = kbase + (size_t)srow * rs;
    const float* frow = freqs + (size_t)srow * AT_D;
#pragma unroll
    for (int t = 0; t < 32; ++t) {
      const int d = t * 4 + kb;
      v2f kv = *(const v2f*)(krow + d);
      const float c = frow[d];
      const float s = frow[d + 1];
      v2f bf;
      bf.x = kv.x * c - kv.y * s;        // K^T[d  ][n]
      bf.y = kv.y * c + kv.x * s;        // K^T[d+1][n]
      sc = wmma_f32(qfrag[t], bf, sc);
    }

    // ---- causal mask on the diagonal block ----
    if (key0 == q0) {
#pragma unroll
      for (int r = 0; r < 8; ++r) {
        const int m = r + half * 8;
        if (l16 > m) sc[r] = -3.4e38f;
      }
    }

    // ---- online softmax (16-lane butterflies stay inside each half) ----
#pragma unroll
    for (int r = 0; r < 8; ++r) {
      float v = sc[r];
      float bm = v;
      bm = fmaxf(bm, __shfl_xor(bm, 1));
      bm = fmaxf(bm, __shfl_xor(bm, 2));
      bm = fmaxf(bm, __shfl_xor(bm, 4));
      bm = fmaxf(bm, __shfl_xor(bm, 8));
      const float nm  = fmaxf(rowmax[r], bm);
      const float scl = __expf(rowmax[r] - nm);
      rowmax[r] = nm;
      const float p = __expf(v - nm);
      sc[r] = p;
      float ps = p;
      ps += __shfl_xor(ps, 1);
      ps += __shfl_xor(ps, 2);
      ps += __shfl_xor(ps, 4);
      ps += __shfl_xor(ps, 8);
      rowsum[r] = rowsum[r] * scl + ps;
#pragma unroll
      for (int nt = 0; nt < 8; ++nt) acc[nt][r] *= scl;
    }

    // ---- stage P (C-layout) through LDS to re-read in A-layout ----
#pragma unroll
    for (int r = 0; r < 8; ++r)
      plds[wv][r + half * 8][l16] = sc[r];
    asm volatile("s_wait_dscnt 0x0" ::: "memory");   // store->load, same wave

    // ---- acc += P @ V  (4 k-steps x 8 n-tiles = 32 WMMAs) ----
#pragma unroll
    for (int t = 0; t < 4; ++t) {
      v2f pa;
      pa.x = plds[wv][l16][t * 4 + kb];
      pa.y = plds[wv][l16][t * 4 + kb + 1];
      const int kr = key0 + t * 4 + kb;
      const float* vrow = vbase + (size_t)kr * rs;
#pragma unroll
      for (int nt = 0; nt < 8; ++nt) {
        v2f bf;
        bf.x = vrow[nt * 16 + l16];      // V[kr  ][d]
        bf.y = vrow[rs + nt * 16 + l16]; // V[kr+1][d]
        acc[nt] = wmma_f32(pa, bf, acc[nt]);
      }
    }
  }

  // ---- epilogue: out[b, q0+m, h*D + d] = acc / rowsum ----
#pragma unroll
  for (int r = 0; r < 8; ++r) {
    const int m = r + half * 8;
    const float inv = 1.0f / rowsum[r];
    float* orow = out + (size_t)(b * AT_S + q0 + m) * AT_E + h * AT_D;
#pragma unroll
    for (int nt = 0; nt < 8; ++nt)
      orow[nt * 16 + l16] = acc[nt][r] * inv;
  }
}

// ---------------------------------------------------------------------------
// Launch: x[2,2048,2048] @ Wqkv[2048,6144] -> qkv ; flash attn ; @ Wo -> out
// ---------------------------------------------------------------------------
extern "C" void kernel_launch(void* const* d_in, const int* in_sizes, int n_in,
                              void* d_out, int out_size, void* d_ws, size_t ws_size,
                              hipStream_t stream) {
  (void)in_sizes; (void)n_in; (void)out_size; (void)ws_size;

  const float* x     = (const float*)d_in[0];   // [B,S,E]   = [2,2048,2048]
  const float* Wqkv  = (const float*)d_in[1];   // [E,3E]    = [2048,6144]
  const float* Wo    = (const float*)d_in[2];   // [E,E]
  const float* freqs = (const float*)d_in[3];   // [S,1,64,2]
  float* out = (float*)d_out;                   // [B,S,E]

  const int BS = 2 * 2048;                      // B*S = 4096 rows
  const int E  = 2048;

  float* qkv  = (float*)d_ws;                   // BS * 3E floats (100.7 MB)
  float* attn = qkv + (size_t)BS * 3 * E;       // BS * E  floats ( 33.6 MB)

  dim3 blk(256);

  // qkv = x @ Wqkv : M=4096, N=6144, K=2048
  gemm_wmma_f32<<<dim3(6144 / 256, BS / 128), blk, 0, stream>>>(
      x, Wqkv, qkv, BS, 3 * E, E);

  // causal flash attention with fused RoPE: 4096 waves, 8 waves/block
  attn_flash_wmma<<<dim3((2 * AT_H * (AT_S / 16)) / 8), blk, 0, stream>>>(
      qkv, freqs, attn);

  // out = attn @ Wo : M=4096, N=2048, K=2048
  gemm_wmma_f32<<<dim3(E / 256, BS / 128), blk, 0, stream>>>(
      attn, Wo, out, BS, E, E);
}